// GMMLoss_73547019977334
// MI455X (gfx1250) — compile-verified
//
#include <hip/hip_runtime.h>
#include <hip/hip_bf16.h>
#include <math.h>

// Problem constants (match reference)
#define BATCH 16384
#define KDIM  256
#define LCLS  10
#define SIGMA_V 1.0f

// Sorted-path geometry: class regions padded to CH rows; chunk = CH rows.
// Chunks are stored ROW-PAIR INTERLEAVED: element (lr,k) of a chunk lives at
// ((lr>>1)*KDIM + k)*2 + (lr&1), so every WMMA operand pair {row2m,row2m+1}
// is one contiguous 8-byte LDS word -> single ds_load_b64, no operand movs.
#define CH      64                 // rows per staged chunk (64 KB LDS)
#define NSPLIT  16                 // batch splits (grid.y)
#define PADTOT  17408              // >= 16384 + 10*(CH-1), NCHUNK % NSPLIT == 0
#define NCHUNK  (PADTOT / CH)      // 272
#define JQ      4                  // j-tiles per wave (A-operand reuse)
static_assert(NCHUNK % NSPLIT == 0, "chunk split");

typedef __attribute__((ext_vector_type(2))) float v2f;
typedef __attribute__((ext_vector_type(8))) float v8f;

// D(16x16 f32) = A(16x4 f32) * B(4x16 f32) + C
__device__ __forceinline__ v8f wmma_f32_16x16x4(v2f a, v2f b, v8f c) {
  return __builtin_amdgcn_wmma_f32_16x16x4_f32(false, a, false, b, (short)0, c,
                                               false, false);
}
__device__ __forceinline__ v8f vzero8() {
  v8f z = {0.f, 0.f, 0.f, 0.f, 0.f, 0.f, 0.f, 0.f};
  return z;
}

// CDNA5 async global->LDS copy (ASYNCcnt path), SADDR form.
__device__ __forceinline__ void async_ld_b128(unsigned lds_off, unsigned goff,
                                              const void* base) {
  asm volatile("global_load_async_to_lds_b128 %0, %1, %2"
               :: "v"(lds_off), "v"(goff), "s"(base) : "memory");
}
__device__ __forceinline__ void wait_async0() {
  asm volatile("s_wait_asynccnt 0" ::: "memory");
}

// ---------------------------------------------------------------- k0: zero
__global__ void k0_zero(float* __restrict__ ws, float* __restrict__ out, int n) {
  int i = blockIdx.x * blockDim.x + threadIdx.x;
  if (i < n) ws[i] = 0.f;
  if (i == 0) out[0] = 0.f;
}
__global__ void k0b_zero(float* __restrict__ p, int n) {
  int i = blockIdx.x * blockDim.x + threadIdx.x;
  if (i < n) p[i] = 0.f;
}

// ------------------------------------------------- k1: class counts & sums
__global__ void k1_count_sums(const float* __restrict__ mu,
                              const int* __restrict__ lab,
                              float* __restrict__ counts,
                              float* __restrict__ sums) {
  __shared__ float lsum[LCLS * KDIM];
  __shared__ float lcnt[LCLS];
  int tid = threadIdx.x;
  for (int c = 0; c < LCLS; ++c) lsum[c * KDIM + tid] = 0.f;
  if (tid < LCLS) lcnt[tid] = 0.f;
  __syncthreads();
  int r0 = blockIdx.x * 256;
  for (int r = r0; r < r0 + 256; ++r) {
    int l = lab[r];
    lsum[l * KDIM + tid] += mu[(size_t)r * KDIM + tid];
    if (tid == 0) lcnt[l] += 1.f;
  }
  __syncthreads();
  for (int c = 0; c < LCLS; ++c)
    atomicAdd(&sums[c * KDIM + tid], lsum[c * KDIM + tid]);
  if (tid < LCLS) atomicAdd(&counts[tid], lcnt[tid]);
}

// ---------------- k_rank: deterministic stable rank per row within class,
//                  padded (to CH) class offsets, per-chunk class table.
__global__ void k_rank(const int* __restrict__ lab, int* __restrict__ pos,
                       int* __restrict__ off, int* __restrict__ chunkCls) {
  __shared__ int cnt[LCLS];
  int t = threadIdx.x;  // one wave
  if (t < LCLS) {
    int c = 0;
    for (int r = 0; r < BATCH; ++r)
      if (lab[r] == t) pos[r] = c++;
    cnt[t] = c;
  }
  __syncthreads();
  if (t == 0) {
    int o = 0;
    for (int l = 0; l < LCLS; ++l) {
      off[l] = o;
      o += ((cnt[l] + CH - 1) / CH) * CH;
    }
    off[LCLS] = o;
    int l = 0;
    for (int c = 0; c < NCHUNK; ++c) {
      int row = c * CH;
      while (l < LCLS && row >= off[l + 1]) ++l;
      chunkCls[c] = (l < LCLS) ? l : 0;  // zero tail rows -> harmless class 0
    }
  }
}

// ---------------- k_sort: bin rows by class into row-pair interleaved chunks
__global__ void k_sort(const float* __restrict__ mu, const int* __restrict__ lab,
                       const int* __restrict__ pos, const int* __restrict__ off,
                       float* __restrict__ sorted) {
  int tid = threadIdx.x, wave = tid >> 5, lane = tid & 31;
  int base = blockIdx.x * 256;  // 64 blocks cover 16384 rows
  for (int it = 0; it < 32; ++it) {
    int r = base + it * 8 + wave;
    int dstRow = off[lab[r]] + pos[r];
    int c  = dstRow / CH;
    int lr = dstRow % CH;
    float* d = sorted + (size_t)c * CH * KDIM + ((size_t)(lr >> 1)) * 0 /*see below*/;
    const float* s = mu + (size_t)r * KDIM;
    size_t chunkBase = (size_t)c * CH * KDIM;
    int par = lr & 1, rp = lr >> 1;
    (void)d;
#pragma unroll
    for (int q = 0; q < 8; ++q) {
      int k = q * 32 + lane;
      sorted[chunkBase + ((size_t)rp * KDIM + k) * 2 + par] = s[k];
    }
  }
}

// ---------------- k2_sorted: pure per-class SYRK, fp32 WMMA, async staging,
//                  1x4 N-blocking; interleaved layout -> b64 operand loads.
__global__ void k2_sorted(const float* __restrict__ sorted,
                          const int* __restrict__ chunkCls,
                          float* __restrict__ Sacc) {
  __shared__ __align__(16) float smu[CH * KDIM];  // 64 KB stage (interleaved)
  int tid = threadIdx.x, wave = tid >> 5, lane = tid & 31;
  int t = blockIdx.x * 8 + wave;                  // tile-quad 0..63
  int i0 = (t >> 2) << 4;                         // 16 row-tiles
  int j0 = (t & 3) << 6;                          // 4 quads of 4 col-tiles
  int p = lane & 15, half = lane >> 4;
  int cbeg = blockIdx.y * (NCHUNK / NSPLIT);
  int cend = cbeg + (NCHUNK / NSPLIT);

  v8f acc[JQ];
#pragma unroll
  for (int q = 0; q < JQ; ++q) acc[q] = vzero8();
  int cur = chunkCls[cbeg];
  unsigned ldsbase = (unsigned)(size_t)smu;

  for (int c = cbeg; c < cend; ++c) {
    int cls = chunkCls[c];            // block-uniform
    if (cls != cur) {                 // flush accumulators on class change
      float* dst = Sacc + (size_t)cur * KDIM * KDIM;
#pragma unroll
      for (int q = 0; q < JQ; ++q)
#pragma unroll
        for (int r = 0; r < 8; ++r)
          atomicAdd(&dst[(size_t)(i0 + r + half * 8) * KDIM + j0 + q * 16 + p],
                    acc[q][r]);
#pragma unroll
      for (int q = 0; q < JQ; ++q) acc[q] = vzero8();
      cur = cls;
    }
    __syncthreads();                  // prior compute done before restage
    const float* src = sorted + (size_t)c * CH * KDIM;
#pragma unroll
    for (int v = 0; v < (CH * KDIM * 4) / (16 * 256); ++v) {  // 16 b128/thread
      unsigned o = (unsigned)(v * 256 + tid) * 16u;
      async_ld_b128(ldsbase + o, o, src);
    }
    wait_async0();
    __syncthreads();                  // all waves' async data visible
#pragma unroll
    for (int s = 0; s < CH / 4; ++s) {
      int rp = s * 2 + half;          // row-pair index: rows s*4+2*half, +1
      const v2f* row = (const v2f*)&smu[(size_t)rp * KDIM * 2];
      v2f a = row[i0 + p];            // single ds_load_b64, even VGPR pair
#pragma unroll
      for (int q = 0; q < JQ; ++q) {  // 5 b64 DS loads per 4 WMMAs, no movs
        v2f b = row[j0 + q * 16 + p];
        acc[q] = wmma_f32_16x16x4(a, b, acc[q]);
      }
    }
  }
  float* dst = Sacc + (size_t)cur * KDIM * KDIM;
#pragma unroll
  for (int q = 0; q < JQ; ++q)
#pragma unroll
    for (int r = 0; r < 8; ++r)
      atomicAdd(&dst[(size_t)(i0 + r + half * 8) * KDIM + j0 + q * 16 + p],
                acc[q][r]);
}

// ---------------- k2_scatter: masked fallback (used only if ws too small)
#define ROWS 32
#define FSPLIT 8
__global__ void k2_scatter(const float* __restrict__ mu,
                           const int* __restrict__ lab,
                           float* __restrict__ Sacc) {
  __shared__ float smu[ROWS * KDIM];
  __shared__ int   slab[ROWS];
  int tid = threadIdx.x, wave = tid >> 5, lane = tid & 31;
  int t = blockIdx.x * 8 + wave;
  int i0 = (t >> 4) << 4, j0 = (t & 15) << 4;
  int p = lane & 15, half = lane >> 4;
  int rbeg = blockIdx.y * (BATCH / FSPLIT);
  int rend = rbeg + (BATCH / FSPLIT);
  v8f acc[LCLS];
#pragma unroll
  for (int c = 0; c < LCLS; ++c) acc[c] = vzero8();
  for (int r0 = rbeg; r0 < rend; r0 += ROWS) {
    __syncthreads();
    const float4* src = (const float4*)(mu + (size_t)r0 * KDIM);
    float4* dst = (float4*)smu;
#pragma unroll
    for (int v = 0; v < (ROWS * KDIM / 4) / 256; ++v)
      dst[v * 256 + tid] = src[v * 256 + tid];
    if (tid < ROWS) slab[tid] = lab[r0 + tid];
    __syncthreads();
    if (r0 + ROWS < rend)
      __builtin_prefetch(mu + (size_t)(r0 + ROWS) * KDIM + tid * 4, 0, 1);
#pragma unroll
    for (int s = 0; s < ROWS / 4; ++s) {
      int rA = s * 4 + half * 2;
      v2f a = { smu[rA * KDIM + i0 + p], smu[(rA + 1) * KDIM + i0 + p] };
      float bxv = smu[rA * KDIM + j0 + p];
      float byv = smu[(rA + 1) * KDIM + j0 + p];
      int l0 = slab[rA], l1 = slab[rA + 1];
#pragma unroll
      for (int c = 0; c < LCLS; ++c) {
        v2f b = { (l0 == c) ? bxv : 0.f, (l1 == c) ? byv : 0.f };
        acc[c] = wmma_f32_16x16x4(a, b, acc[c]);
      }
    }
  }
#pragma unroll
  for (int c = 0; c < LCLS; ++c)
#pragma unroll
    for (int r = 0; r < 8; ++r)
      atomicAdd(&Sacc[(size_t)c * KDIM * KDIM +
                      (size_t)(i0 + r + half * 8) * KDIM + j0 + p],
                acc[c][r]);
}

// -------------------------------- k3: S = A/n - mu_c mu_c^T + sigma*I
__global__ void k3_finalize(float* __restrict__ S,
                            const float* __restrict__ counts,
                            const float* __restrict__ sums) {
  size_t idx = (size_t)blockIdx.x * 256 + threadIdx.x;
  int l = (int)(idx >> 16);
  int rem = (int)(idx & 65535);
  int i = rem >> 8, j = rem & 255;
  float safe = fmaxf(counts[l], 1.f);
  float mi = sums[l * KDIM + i] / safe;
  float mj = sums[l * KDIM + j] / safe;
  S[idx] = S[idx] / safe - mi * mj + ((i == j) ? SIGMA_V : 0.f);
}

// ----------------- k4a: Cholesky of 256x256 SPD matrix in 256KB LDS
__global__ void k4a_cholesky(const float* __restrict__ S,
                             float* __restrict__ chol,
                             float* __restrict__ logdet) {
  extern __shared__ float M[];
  __shared__ float red[256];
  int tid = threadIdx.x, cls = blockIdx.x;
  const float* Sg = S + (size_t)cls * KDIM * KDIM;
  for (int v = 0; v < KDIM; ++v) M[v * KDIM + tid] = Sg[(size_t)v * KDIM + tid];
  __syncthreads();
  for (int j = 0; j < KDIM; ++j) {
    if (tid == 0) M[j * KDIM + j] = sqrtf(M[j * KDIM + j]);
    __syncthreads();
    float dj = M[j * KDIM + j];
    if (tid > j) M[tid * KDIM + j] /= dj;
    __syncthreads();
    if (tid > j) {
      float lij = M[tid * KDIM + j];
      for (int k = j + 1; k <= tid; ++k)
        M[tid * KDIM + k] -= lij * M[k * KDIM + j];
    }
    __syncthreads();
  }
  red[tid] = 2.f * logf(M[tid * KDIM + tid]);
  __syncthreads();
  for (int s = 128; s > 0; s >>= 1) {
    if (tid < s) red[tid] += red[tid + s];
    __syncthreads();
  }
  if (tid == 0) logdet[cls] = red[0];
  float* Cg = chol + (size_t)cls * KDIM * KDIM;
  for (int v = 0; v < KDIM; ++v)
    Cg[(size_t)v * KDIM + tid] = (tid <= v) ? M[v * KDIM + tid] : 0.f;
}

// ----------------- k4b: X = L^-1 in LDS, overwrites chol buffer
__global__ void k4b_trinv(float* __restrict__ chol) {
  extern __shared__ float X[];
  int c = threadIdx.x, cls = blockIdx.x;
  const float* Lg = chol + (size_t)cls * KDIM * KDIM;
  for (int r = 0; r < c; ++r) X[r * KDIM + c] = 0.f;
  X[c * KDIM + c] = 1.f / Lg[(size_t)c * KDIM + c];
  for (int r = c + 1; r < KDIM; ++r) {
    float s = 0.f;
    for (int k = c; k < r; ++k)
      s += Lg[(size_t)r * KDIM + k] * X[k * KDIM + c];
    X[r * KDIM + c] = -s / Lg[(size_t)r * KDIM + r];
  }
  __syncthreads();
  float* Xg = chol + (size_t)cls * KDIM * KDIM;
  for (int v = 0; v < KDIM; ++v) Xg[(size_t)v * KDIM + c] = X[v * KDIM + c];
}

// ----------------- k4c: Sinv = X^T X via fp32 WMMA, X in 256KB LDS
__global__ void k4c_syrk(const float* __restrict__ Xbuf,
                         float* __restrict__ Sinv) {
  extern __shared__ float Xs[];
  int tid = threadIdx.x, wave = tid >> 5, lane = tid & 31;
  int p = lane & 15, half = lane >> 4;
  int cls = blockIdx.x;
  const float4* src = (const float4*)(Xbuf + (size_t)cls * KDIM * KDIM);
  float4* dst = (float4*)Xs;
  for (int v = 0; v < 64; ++v) dst[v * 256 + tid] = src[v * 256 + tid];
  __syncthreads();
  float* out = Sinv + (size_t)cls * KDIM * KDIM;
  for (int t = wave; t < 256; t += 8) {
    int i0 = (t >> 4) << 4, j0 = (t & 15) << 4;
    v8f acc = vzero8();
    for (int kb = 0; kb < KDIM; kb += 4) {
      int rA = kb + half * 2;
      v2f a = { Xs[rA * KDIM + i0 + p], Xs[(rA + 1) * KDIM + i0 + p] };
      v2f b = { Xs[rA * KDIM + j0 + p], Xs[(rA + 1) * KDIM + j0 + p] };
      acc = wmma_f32_16x16x4(a, b, acc);
    }
#pragma unroll
    for (int r = 0; r < 8; ++r)
      out[(size_t)(i0 + r + half * 8) * KDIM + j0 + p] = acc[r];
  }
}

// ----------------- k5: per (i,j): trace, mahalanobis, KL, weighted sum
__global__ void k5_pairs(const float* __restrict__ S,
                         const float* __restrict__ Sinv,
                         const float* __restrict__ counts,
                         const float* __restrict__ sums,
                         const float* __restrict__ logdet,
                         float* __restrict__ out) {
  __shared__ float d[KDIM];
  __shared__ float redT[256], redM[256];
  int tid = threadIdx.x;
  int i = blockIdx.x / LCLS, j = blockIdx.x % LCLS;
  float ni = counts[i], nj = counts[j];
  float si = fmaxf(ni, 1.f), sj = fmaxf(nj, 1.f);
  d[tid] = sums[j * KDIM + tid] / sj - sums[i * KDIM + tid] / si;
  __syncthreads();
  const float* Sj = Sinv + (size_t)j * KDIM * KDIM;
  const float* Si = S + (size_t)i * KDIM * KDIM;
  float tA = 0.f, mA = 0.f;
  for (int idx = tid; idx < KDIM * KDIM; idx += 256) {
    int a = idx >> 8, b = idx & 255;
    float sv = Sj[idx];
    tA += sv * Si[b * KDIM + a];
    mA += d[a] * sv * d[b];
  }
  redT[tid] = tA; redM[tid] = mA;
  __syncthreads();
  for (int s = 128; s > 0; s >>= 1) {
    if (tid < s) { redT[tid] += redT[tid + s]; redM[tid] += redM[tid + s]; }
    __syncthreads();
  }
  if (tid == 0 && i != j) {
    float kl = 0.5f * (logdet[j] - logdet[i] - (float)KDIM + redT[0] + redM[0]);
    double w = (double)ni * (double)nj;
    const double invB3 = 1.0 / ((double)BATCH * (double)BATCH * (double)BATCH);
    atomicAdd(out, (float)((double)kl * w * invB3));
  }
}

extern "C" void kernel_launch(void* const* d_in, const int* in_sizes, int n_in,
                              void* d_out, int out_size, void* d_ws, size_t ws_size,
                              hipStream_t stream) {
  (void)in_sizes; (void)n_in; (void)out_size;
  const float* mu  = (const float*)d_in[0];
  const int*   lab = (const int*)d_in[1];
  float* out = (float*)d_out;
  float* ws  = (float*)d_ws;

  // Workspace layout (float units)
  const size_t MSZ = (size_t)LCLS * KDIM * KDIM;         // 655360
  float* counts = ws;                                    // 16
  float* sums   = ws + 16;                               // L*K
  float* Smat   = sums + LCLS * KDIM;                    // A -> S in place
  float* chol   = Smat + MSZ;                            // L, then X=L^-1
  float* Sinv   = chol + MSZ;
  float* logdet = Sinv + MSZ;                            // 16
  size_t fbase  = 16 + (size_t)LCLS * KDIM + 3 * MSZ + 16;
  int*   off      = (int*)(ws + fbase);                  // 16
  int*   chunkCls = off + 16;                            // 512 (272 used)
  int*   pos      = chunkCls + 512;                      // BATCH
  float* sorted   = ws + fbase + 16 + 512 + BATCH;       // PADTOT*KDIM
  size_t needBytes = (fbase + 16 + 512 + BATCH + (size_t)PADTOT * KDIM) * 4;

  int nzero = (int)(16 + LCLS * KDIM + MSZ);
  k0_zero<<<(nzero + 255) / 256, 256, 0, stream>>>(ws, out, nzero);
  k1_count_sums<<<BATCH / 256, 256, 0, stream>>>(mu, lab, counts, sums);

  if (ws_size >= needBytes) {
    // Sorted path: bin rows by class, then pure per-class WMMA SYRK.
    int nsz = PADTOT * KDIM;
    k0b_zero<<<(nsz + 255) / 256, 256, 0, stream>>>(sorted, nsz);
    k_rank<<<1, 32, 0, stream>>>(lab, pos, off, chunkCls);
    k_sort<<<BATCH / 256, 256, 0, stream>>>(mu, lab, pos, off, sorted);
    k2_sorted<<<dim3(8, NSPLIT), 256, 0, stream>>>(sorted, chunkCls, Smat);
  } else {
    // Fallback: masked segmented SYRK (10x algebraic work, no extra ws).
    k2_scatter<<<dim3(32, FSPLIT), 256, 0, stream>>>(mu, lab, Smat);
  }

  k3_finalize<<<(unsigned)(MSZ / 256), 256, 0, stream>>>(Smat, counts, sums);
  k4a_cholesky<<<LCLS, 256, KDIM * KDIM * sizeof(float), stream>>>(Smat, chol, logdet);
  k4b_trinv<<<LCLS, 256, KDIM * KDIM * sizeof(float), stream>>>(chol);
  k4c_syrk<<<LCLS, 256, KDIM * KDIM * sizeof(float), stream>>>(chol, Sinv);
  k5_pairs<<<LCLS * LCLS, 256, 0, stream>>>(Smat, Sinv, counts, sums, logdet, out);
}